// PConv_20255065768439
// MI455X (gfx1250) — compile-verified
//
#include <hip/hip_runtime.h>

typedef __attribute__((ext_vector_type(2))) float v2f;
typedef __attribute__((ext_vector_type(8))) float v8f;

#define B_      4
#define N_      20000
#define M_      20000
#define K_      16
#define C_IN    64
#define C_MID   16
#define C_ADD   3
#define C_TOT   (C_IN + C_ADD)        // 67
#define OUT_PP  (C_TOT * C_MID)       // 1072
#define WAVES_PER_BLOCK 8

// One wave32 per output point.  D(67x16) = F^T(67x16) x W(16x16), built from
// 5 C-tiles x 4 K-steps of v_wmma_f32_16x16x4_f32.
__global__ __launch_bounds__(256) void pconv_wmma_kernel(
    const float* __restrict__ feats,   // [B, N, 64]
    const int*   __restrict__ inds,    // [B, M, 16]
    const float* __restrict__ wnet,    // [B, M, 16, 16]
    const float* __restrict__ addf,    // [B, M, 16, 3]
    float*       __restrict__ out)     // [B, M, 1072]
{
    const int lane = threadIdx.x & 31;
    const int wave = threadIdx.x >> 5;
    const int pt   = blockIdx.x * WAVES_PER_BLOCK + wave;   // exact, no tail

    const int b    = pt / M_;
    const int half = lane >> 4;     // 0 = lanes 0-15, 1 = lanes 16-31
    const int col  = lane & 15;

    const float* fb = feats + (size_t)b  * N_ * C_IN;
    const int*   ip = inds  + (size_t)pt * K_;
    const float* wp = wnet  + (size_t)pt * (K_ * C_MID);
    const float* ap = addf  + (size_t)pt * (K_ * C_ADD);
    float*       op = out   + (size_t)pt * OUT_PP;

    v8f acc[5] = {};   // 5 C-tiles of 16x16 f32 accumulators (c rows x w cols)

    #pragma unroll
    for (int t = 0; t < 4; ++t) {
        // K-slice for this WMMA step: this half-wave owns neighbors kA, kA+1.
        const int kA = t * 4 + half * 2;
        const int n0 = ip[kA];
        const int n1 = ip[kA + 1];

        // B fragment: W rows kA / kA+1, 16 weights wide.  Streaming data ->
        // non-temporal (read exactly once across the whole dispatch).
        v2f bf;
        bf.x = __builtin_nontemporal_load(wp + kA       * C_MID + col);
        bf.y = __builtin_nontemporal_load(wp + (kA + 1) * C_MID + col);

        const float* r0 = fb + (size_t)n0 * C_IN;
        const float* r1 = fb + (size_t)n1 * C_IN;

        // C-tiles 0..3: gathered input features (L2-resident, temporal loads;
        // each half-wave reads 16 contiguous floats -> coalesced 64B).
        #pragma unroll
        for (int ct = 0; ct < 4; ++ct) {
            v2f af;
            af.x = r0[ct * 16 + col];
            af.y = r1[ct * 16 + col];
            acc[ct] = __builtin_amdgcn_wmma_f32_16x16x4_f32(
                false, af, false, bf, (short)0, acc[ct], false, false);
        }

        // C-tile 4: channels 64..66 from additional_features, 67..79 zero.
        // Selects (not branches) keep EXEC all-ones for the WMMA.
        {
            v2f af;
            af.x = (col < C_ADD)
                 ? __builtin_nontemporal_load(ap + kA       * C_ADD + col) : 0.f;
            af.y = (col < C_ADD)
                 ? __builtin_nontemporal_load(ap + (kA + 1) * C_ADD + col) : 0.f;
            acc[4] = __builtin_amdgcn_wmma_f32_16x16x4_f32(
                false, af, false, bf, (short)0, acc[4], false, false);
        }
    }

    // Store: D layout -> VGPR i, lane l<16: out[(cbase+i)*16 + l];
    // lanes 16-31 carry rows cbase+8..cbase+15.  343MB streaming output ->
    // non-temporal stores so L2 stays dedicated to the gather table.
    #pragma unroll
    for (int ct = 0; ct < 4; ++ct) {
        const int cbase = ct * 16 + half * 8;
        #pragma unroll
        for (int i = 0; i < 8; ++i) {
            __builtin_nontemporal_store(acc[ct][i], op + (cbase + i) * C_MID + col);
        }
    }
    // Tail tile: only rows c = 64..66 are real (half 0, i < 3).
    if (half == 0) {
        #pragma unroll
        for (int i = 0; i < C_ADD; ++i) {
            __builtin_nontemporal_store(acc[4][i], op + (C_IN + i) * C_MID + col);
        }
    }
}

extern "C" void kernel_launch(void* const* d_in, const int* in_sizes, int n_in,
                              void* d_out, int out_size, void* d_ws, size_t ws_size,
                              hipStream_t stream) {
    const float* feats = (const float*)d_in[0];   // [B,N,64]  f32
    const int*   inds  = (const int*)  d_in[1];   // [B,M,16]  int
    const float* wnet  = (const float*)d_in[2];   // [B,M,16,16] f32
    const float* addf  = (const float*)d_in[3];   // [B,M,16,3]  f32
    float*       outp  = (float*)d_out;           // [B,M,1072] f32

    const int total_pts = B_ * M_;                    // 80000
    const int blocks    = total_pts / WAVES_PER_BLOCK; // 10000, exact

    hipLaunchKernelGGL(pconv_wmma_kernel, dim3(blocks), dim3(256), 0, stream,
                       feats, inds, wnet, addf, outp);
}